// LRUReducedLinear_77043123356067
// MI455X (gfx1250) — compile-verified
//
#include <hip/hip_runtime.h>
#include <hip/hip_bf16.h>

#define IN      11008
#define OUTD    4096
#define TOPK    2752
#define RECALL_TH 2201
#define CACHE   16
#define BSZ     8
#define KSPLIT  8

typedef __attribute__((ext_vector_type(2))) float v2f;
typedef __attribute__((ext_vector_type(8))) float v8f;

// ---------------- kernel 1: scores[i] = sum_b |x[b,i]| ----------------
__global__ void k_scores(const float* __restrict__ x, float* __restrict__ scores) {
    int i = blockIdx.x * blockDim.x + threadIdx.x;
    if (i < IN) {
        float s = 0.f;
#pragma unroll
        for (int b = 0; b < BSZ; ++b) s += fabsf(x[b * IN + i]);
        scores[i] = s;
    }
}

// ---------------- kernel 2: top-k selection -> mask (exact, tie = lowest idx) ----
// scores >= 0, so uint bit pattern order == float order.
__global__ void k_select(const float* __restrict__ scores, int* __restrict__ mask) {
    __shared__ int red[1024];
    const int tid = threadIdx.x;

    unsigned lo = 0u, hi = 0x80000000u;   // cnt(lo)>=TOPK, cnt(hi)<TOPK invariant
#pragma unroll 1
    for (int it = 0; it < 32; ++it) {
        unsigned mid = lo + ((hi - lo) >> 1);
        int c = 0;
        for (int i = tid; i < IN; i += 1024)
            c += (__float_as_uint(scores[i]) >= mid) ? 1 : 0;
        red[tid] = c; __syncthreads();
        for (int s = 512; s > 0; s >>= 1) {
            if (tid < s) red[tid] += red[tid + s];
            __syncthreads();
        }
        int total = red[0];
        __syncthreads();
        if (total >= TOPK) lo = mid; else hi = mid;
    }
    const unsigned T = lo;  // bit pattern of the TOPK-th largest score

    // count strictly-greater
    {
        int c = 0;
        for (int i = tid; i < IN; i += 1024)
            c += (__float_as_uint(scores[i]) > T) ? 1 : 0;
        red[tid] = c; __syncthreads();
        for (int s = 512; s > 0; s >>= 1) {
            if (tid < s) red[tid] += red[tid + s];
            __syncthreads();
        }
    }
    const int count_gt = red[0];
    __syncthreads();

    for (int i = tid; i < IN; i += 1024)
        mask[i] = (__float_as_uint(scores[i]) > T) ? 1 : 0;
    __syncthreads();

    if (tid == 0) {   // admit ties in ascending index order (matches lax.top_k)
        int need = TOPK - count_gt;
        for (int i = 0; i < IN && need > 0; ++i) {
            if (__float_as_uint(scores[i]) == T) { mask[i] = 1; --need; }
        }
    }
}

// ---------------- kernel 3: recall[c] = popcount(mask & cached_mask[c]) -------
__global__ void k_recall(const int* __restrict__ mask,
                         const unsigned char* __restrict__ cmask,
                         int* __restrict__ recall) {
    __shared__ int red[256];
    const int c = blockIdx.x, t = threadIdx.x;
    int s = 0;
    for (int i = t; i < IN; i += 256)
        s += (mask[i] && cmask[(size_t)c * IN + i]) ? 1 : 0;
    red[t] = s; __syncthreads();
    for (int k = 128; k > 0; k >>= 1) {
        if (t < k) red[t] += red[t + k];
        __syncthreads();
    }
    if (t == 0) recall[c] = red[0];
}

// ---------------- kernel 4: argmax + write decision ----------------
__global__ void k_decide(const int* __restrict__ recall, int* __restrict__ meta) {
    if (threadIdx.x == 0 && blockIdx.x == 0) {
        int best = recall[0], bi = 0;
        for (int c = 1; c < CACHE; ++c)
            if (recall[c] > best) { best = recall[c]; bi = c; }   // first max on ties
        meta[0] = (best < RECALL_TH) ? 1 : 0;  // write flag
        meta[1] = bi;                          // max_idx
    }
}

// ---------------- kernel 5: build padded A matrices (16 x K) ----------------
// XpA (K=IN):   masked dense x, rows 8..15 zero  (write path)
// XpB (K=TOPK): x gathered by cached_indices[max_idx], rows 8..15 zero (cached path)
__global__ void k_prep(const float* __restrict__ x,
                       const int* __restrict__ mask,
                       const int* __restrict__ cidx,
                       const int* __restrict__ meta,
                       float* __restrict__ XpA, float* __restrict__ XpB) {
    int g = blockIdx.x * blockDim.x + threadIdx.x;
    const int NA = 16 * IN;
    if (g < NA) {
        int m = g / IN, k = g - m * IN;
        XpA[g] = (m < BSZ && mask[k]) ? x[m * IN + k] : 0.f;
    } else {
        int h = g - NA;
        if (h < 16 * TOPK) {
            int m = h / TOPK, k = h - m * TOPK;
            int idx = cidx[(size_t)meta[1] * TOPK + k];
            XpB[h] = (m < BSZ) ? x[m * IN + idx] : 0.f;
        }
    }
}

// ---------------- kernel 6: WMMA GEMM partials -------------------------------
// partial[s](8x4096) = A(16 x Kchunk_s) * B(4096 x Kchunk_s)^T
// grid = (64 tiles-of-4-waves, KSPLIT).  One wave per 16-wide N tile per k-chunk.
// V_WMMA_F32_16X16X4_F32; weight stream loaded non-temporally.
__global__ __launch_bounds__(128) void k_gemm(const float* __restrict__ weight,
                                              const float* __restrict__ cached_weight,
                                              const float* __restrict__ XpA,
                                              const float* __restrict__ XpB,
                                              const int* __restrict__ meta,
                                              float* __restrict__ partial) {
    const int lane = threadIdx.x & 31;
    const int wave = threadIdx.x >> 5;
    const int n0   = (blockIdx.x * 4 + wave) * 16;
    const int s    = blockIdx.y;

    // uniform path select (device-side decision; no host readback under graph capture)
    const int write = meta[0];
    const float* A; const float* B; int K;
    if (write) { A = XpA; B = weight;                                        K = IN;   }
    else       { A = XpB; B = cached_weight + (size_t)meta[1] * OUTD * TOPK; K = TOPK; }

    // k-chunk for this block, multiple of 16 (11008 -> 1376; 2752 -> 352/288 tail)
    const int CH   = (((K + KSPLIT - 1) / KSPLIT) + 15) & ~15;
    const int kbeg = s * CH;
    const int kend = min(K, kbeg + CH);

    // fragment addressing per ISA layout:
    //  A 16x4 f32: lanes 0-15 -> (M=lane, K=k,k+1); lanes 16-31 -> (M=lane-16, K=k+2,k+3)
    //  B 4x16     : symmetric => lane reads row n0+(lane&15) at k + 2*(lane>>4)
    const int mrow = lane & 15;
    const int koff = (lane >> 4) << 1;
    const float* aptr = A + (size_t)mrow * K + koff;
    const float* bptr = B + (size_t)(n0 + mrow) * K + koff;

    v8f acc = {0.f, 0.f, 0.f, 0.f, 0.f, 0.f, 0.f, 0.f};

#pragma unroll 1
    for (int k = kbeg; k < kend; k += 16) {
        __builtin_prefetch(bptr + k + 1024, 0, 3);   // stream-ahead on the big W rows
#pragma unroll
        for (int q = 0; q < 16; q += 4) {
            v2f a = *(const v2f*)(aptr + k + q);
            v2f b = __builtin_nontemporal_load((const v2f*)(bptr + k + q));
            acc = __builtin_amdgcn_wmma_f32_16x16x4_f32(
                false, a, false, b, (short)0, acc, false, false);
        }
    }

    // D layout: VGPR r, lanes 0-15 -> (M=r, N=lane). Rows 8..15 are padding.
    if (lane < 16) {
        const int n = n0 + lane;
        float* p = partial + (size_t)s * BSZ * OUTD;
#pragma unroll
        for (int r = 0; r < BSZ; ++r)
            p[(size_t)r * OUTD + n] = acc[r];
    }
}

// ---------------- kernel 7: deterministic reduction + bias -------------------
__global__ void k_final(const float* __restrict__ partial,
                        const float* __restrict__ bias,
                        float* __restrict__ out) {
    int i = blockIdx.x * blockDim.x + threadIdx.x;   // i in [0, 8*4096)
    if (i < BSZ * OUTD) {
        float acc = bias[i & (OUTD - 1)];
#pragma unroll
        for (int s = 0; s < KSPLIT; ++s)
            acc += partial[(size_t)s * BSZ * OUTD + i];
        out[i] = acc;
    }
}

// ---------------- launch ----------------
extern "C" void kernel_launch(void* const* d_in, const int* in_sizes, int n_in,
                              void* d_out, int out_size, void* d_ws, size_t ws_size,
                              hipStream_t stream) {
    const float*         x       = (const float*)d_in[0];
    const float*         weight  = (const float*)d_in[1];
    const float*         bias    = (const float*)d_in[2];
    const float*         cweight = (const float*)d_in[3];
    const unsigned char* cmask   = (const unsigned char*)d_in[4];
    const int*           cidx    = (const int*)d_in[5];
    float* out = (float*)d_out;

    char* ws = (char*)d_ws;
    float* scores  = (float*)(ws);                      // IN floats          44032 B
    int*   mask    = (int*)  (ws + 44032);              // IN ints            44032 B
    int*   recall  = (int*)  (ws + 88064);              // 16 ints
    int*   meta    = (int*)  (ws + 88128);              // flag, max_idx
    float* XpA     = (float*)(ws + 88192);              // 16*IN floats      704512 B
    float* XpB     = (float*)(ws + 792704);             // 16*TOPK floats    176128 B
    float* partial = (float*)(ws + 968832);             // KSPLIT*8*4096 f  1048576 B

    k_scores <<<(IN + 255) / 256, 256, 0, stream>>>(x, scores);
    k_select <<<1, 1024, 0, stream>>>(scores, mask);
    k_recall <<<CACHE, 256, 0, stream>>>(mask, cmask, recall);
    k_decide <<<1, 32, 0, stream>>>(recall, meta);
    {
        int total = 16 * IN + 16 * TOPK;                // 220160 = 860 * 256
        k_prep <<<(total + 255) / 256, 256, 0, stream>>>(x, mask, cidx, meta, XpA, XpB);
    }
    k_gemm <<<dim3(OUTD / 64, KSPLIT), 128, 0, stream>>>(weight, cweight, XpA, XpB, meta, partial);
    k_final <<<(BSZ * OUTD + 255) / 256, 256, 0, stream>>>(partial, bias, out);
}